// SelfAttentionV2_58093727646120
// MI455X (gfx1250) — compile-verified
//
#include <hip/hip_runtime.h>

// Problem constants (from reference)
#define B_   8
#define S_   4096
#define HID_ 1024
#define H_   16
#define DH_  64

typedef __attribute__((ext_vector_type(2))) float v2f;
typedef __attribute__((ext_vector_type(8))) float v8f;

// ---------------- workspace layout (in floats) ----------------
#define OFF_WQE 0                          // 16*1024
#define OFF_WKE (OFF_WQE + H_*HID_)        // 16*1024
#define OFF_QS  (OFF_WKE + H_*HID_)        // B*S*H = 524288 (masked q_scores)
#define OFF_HSP (OFF_QS  + B_*S_*H_)       // B*32*HID = 262144 (hsum partials)
#define OFF_KHP (OFF_HSP + B_*32*HID_)     // 256*16*1024 = 4194304 (kh partials)
#define OFF_HS  (OFF_KHP + 256*H_*HID_)    // B*HID = 8192
#define OFF_KH  (OFF_HS  + B_*HID_)        // B*H*HID = 131072
#define OFF_MVS (OFF_KH  + B_*H_*HID_)     // B*H*DH = 8192 (mean_value/len)
#define OFF_KVS (OFF_MVS + B_*H_*DH_)      // B*H*DH = 8192 (key_values/len)
#define OFF_IL  (OFF_KVS + B_*H_*DH_)      // B (1/length)

// ---------------------------------------------------------------
// Kernel 0: fold att_weight into Wq / Wk:  weff[h,i] = sum_d aw[h,d]*W[h*64+d, i]
__global__ __launch_bounds__(256) void weff_kernel(
    const float* __restrict__ Wq, const float* __restrict__ Wk,
    const float* __restrict__ aw, float* __restrict__ wqe, float* __restrict__ wke)
{
    int u = blockIdx.x * 256 + threadIdx.x;      // 64 blocks -> 16384 = H*HID
    int h = u >> 10, i = u & (HID_ - 1);
    float aq = 0.f, ak = 0.f;
#pragma unroll 4
    for (int d = 0; d < DH_; ++d) {
        float a = aw[h * DH_ + d];
        aq += a * Wq[(size_t)(h * DH_ + d) * HID_ + i];
        ak += a * Wk[(size_t)(h * DH_ + d) * HID_ + i];
    }
    wqe[h * HID_ + i] = aq;
    wke[h * HID_ + i] = ak;
}

// Kernel 1: invlen[b] = 1 / sum_s mask[b,s]
__global__ __launch_bounds__(256) void invlen_kernel(
    const float* __restrict__ mask, float* __restrict__ il)
{
    __shared__ float red[256];
    int b = blockIdx.x, t = threadIdx.x;
    float s = 0.f;
#pragma unroll
    for (int k = 0; k < S_ / 256; ++k) s += mask[b * S_ + t + 256 * k];
    red[t] = s;
    __syncthreads();
    for (int w = 128; w > 0; w >>= 1) {
        if (t < w) red[t] += red[t + w];
        __syncthreads();
    }
    if (t == 0) il[b] = 1.0f / red[0];
}

// ---------------------------------------------------------------
// Kernel 2: main streaming pass. 256 blocks (8 batches x 32 row-tiles of 128).
// Phase A: q/k scores via v_wmma_f32_16x16x4_f32, hsum partials.
// Phase C: kh[h,i] += ks^T @ hidden via WMMA, per-block partials.
__global__ __launch_bounds__(256) void pass1_kernel(
    const float* __restrict__ hidden, const float* __restrict__ mask,
    const float* __restrict__ wqe, const float* __restrict__ wke,
    float* __restrict__ q_scores, float* __restrict__ hsum_part,
    float* __restrict__ kh_part)
{
    __shared__ float lds_tile[128 * 68];   // hidden tile, stride 68 (conflict-free)
    __shared__ float lds_wq[16 * 68];
    __shared__ float lds_wk[16 * 68];
    __shared__ float lds_ksT[16 * 132];    // masked k-scores, [h][s_local], stride 132
    __shared__ float lds_red[256 * 4];     // hsum partial reduce buffer
    __shared__ float lds_khb[2 * 16 * 68]; // kh per-chunk [s-half][h][i_local]

    const int blk = blockIdx.x & 31;
    const int b   = blockIdx.x >> 5;
    const int s0  = blk * 128;
    const int t = threadIdx.x;
    const int wave = t >> 5, lane = t & 31;
    const int lane16 = lane & 15, laneHi = lane >> 4;

    const float* hbase = hidden + ((size_t)b * S_ + s0) * HID_;

    v8f cq = {}; v8f ck = {};

    // ---------- Phase A: scores GEMM + hsum ----------
    for (int kc = 0; kc < 16; ++kc) {
        const int k0 = kc * 64;
        // stage weff chunks (16 x 64 each): one float4 per thread per matrix
        {
            int r = t >> 4, c4 = (t & 15) * 4;
            *(float4*)&lds_wq[r * 68 + c4] = *(const float4*)(wqe + r * HID_ + k0 + c4);
            *(float4*)&lds_wk[r * 68 + c4] = *(const float4*)(wke + r * HID_ + k0 + c4);
        }
        // stage hidden tile (128 x 64) coalesced, accumulate masked col-sums
        float4 acc = {0.f, 0.f, 0.f, 0.f};
#pragma unroll
        for (int it = 0; it < 8; ++it) {
            int row = it * 16 + (t >> 4);
            float4 v = *(const float4*)(hbase + (size_t)row * HID_ + k0 + (t & 15) * 4);
            *(float4*)&lds_tile[row * 68 + (t & 15) * 4] = v;
            float m = mask[b * S_ + s0 + row];
            acc.x += m * v.x; acc.y += m * v.y; acc.z += m * v.z; acc.w += m * v.w;
        }
        *(float4*)&lds_red[t * 4] = acc;
        __syncthreads();
        // reduce 16 partials per column group, write exclusive per-block partial
        if (t < 64) {
            int g = t >> 2, comp = t & 3;
            float s = 0.f;
#pragma unroll
            for (int p = 0; p < 16; ++p) s += lds_red[(p * 16 + g) * 4 + comp];
            hsum_part[(size_t)(b * 32 + blk) * HID_ + k0 + g * 4 + comp] = s;
        }
        // WMMA chain over this K-chunk
        const float* atile = &lds_tile[(wave * 16 + lane16) * 68 + 2 * laneHi];
        const float* bq = &lds_wq[lane16 * 68 + 2 * laneHi];
        const float* bk = &lds_wk[lane16 * 68 + 2 * laneHi];
#pragma unroll
        for (int kk = 0; kk < 16; ++kk) {
            v2f a = *(const v2f*)(atile + kk * 4);
            v2f q = *(const v2f*)(bq + kk * 4);
            v2f k = *(const v2f*)(bk + kk * 4);
            cq = __builtin_amdgcn_wmma_f32_16x16x4_f32(false, a, false, q, (short)0, cq, false, false);
            ck = __builtin_amdgcn_wmma_f32_16x16x4_f32(false, a, false, k, (short)0, ck, false, false);
        }
        __syncthreads();
    }

    // ---------- store masked q_scores, stash masked k-scores^T to LDS ----------
#pragma unroll
    for (int j = 0; j < 8; ++j) {
        int sl = wave * 16 + j + 8 * laneHi;          // local row in [0,128)
        float m = mask[b * S_ + s0 + sl];
        q_scores[((size_t)b * S_ + s0 + sl) * H_ + lane16] = cq[j] * m;
        lds_ksT[lane16 * 132 + sl] = ck[j] * m;
    }
    __syncthreads();

    // ---------- Phase C: kh[h,i] += ksT(16x128) @ hidden(128xHID) ----------
    const int ic = wave & 3;   // i-chunk within 64-col K-chunk
    const int sh = wave >> 2;  // s-half (rows sh*64 .. +63)
    for (int kc = 0; kc < 16; ++kc) {
        const int k0 = kc * 64;
#pragma unroll
        for (int it = 0; it < 8; ++it) {
            int row = it * 16 + (t >> 4);
            *(float4*)&lds_tile[row * 68 + (t & 15) * 4] =
                *(const float4*)(hbase + (size_t)row * HID_ + k0 + (t & 15) * 4);
        }
        __syncthreads();
        v8f ckh = {};
        const float* aks = &lds_ksT[lane16 * 132 + sh * 64 + 2 * laneHi];
#pragma unroll
        for (int ss = 0; ss < 16; ++ss) {
            v2f a = *(const v2f*)(aks + ss * 4);
            int srow = sh * 64 + ss * 4 + 2 * laneHi;
            v2f bb;
            bb.x = lds_tile[srow * 68 + ic * 16 + lane16];
            bb.y = lds_tile[(srow + 1) * 68 + ic * 16 + lane16];
            ckh = __builtin_amdgcn_wmma_f32_16x16x4_f32(false, a, false, bb, (short)0, ckh, false, false);
        }
        // combine the two s-halves through LDS (commutative 2-way add: deterministic)
#pragma unroll
        for (int j = 0; j < 8; ++j)
            lds_khb[(sh * 16 + j + 8 * laneHi) * 68 + ic * 16 + lane16] = ckh[j];
        __syncthreads();
        {
            int h = t >> 4, c4 = (t & 15) * 4;
            float4 a0 = *(const float4*)&lds_khb[h * 68 + c4];
            float4 a1 = *(const float4*)&lds_khb[(16 + h) * 68 + c4];
            float4 r; r.x = a0.x + a1.x; r.y = a0.y + a1.y; r.z = a0.z + a1.z; r.w = a0.w + a1.w;
            *(float4*)&kh_part[(((size_t)(b * 32 + blk)) * H_ + h) * HID_ + k0 + c4] = r;
        }
        __syncthreads();
    }
}

// ---------------------------------------------------------------
// Kernel 3a/3b: fixed-order reduction of per-block partials
__global__ __launch_bounds__(256) void reduce_hsum_kernel(
    const float* __restrict__ hsp, float* __restrict__ hs)
{
    int idx = blockIdx.x * 256 + threadIdx.x;       // 32 blocks -> B*HID
    int b = idx >> 10, i = idx & (HID_ - 1);
    float s = 0.f;
#pragma unroll 8
    for (int blk = 0; blk < 32; ++blk) s += hsp[(size_t)(b * 32 + blk) * HID_ + i];
    hs[idx] = s;
}

__global__ __launch_bounds__(256) void reduce_kh_kernel(
    const float* __restrict__ khp, float* __restrict__ kh)
{
    int idx = blockIdx.x * 256 + threadIdx.x;       // 512 blocks -> B*H*HID
    int bh = idx >> 10, i = idx & (HID_ - 1);
    int b = bh >> 4, h = bh & 15;
    float s = 0.f;
#pragma unroll 8
    for (int blk = 0; blk < 32; ++blk)
        s += khp[(((size_t)(b * 32 + blk)) * H_ + h) * HID_ + i];
    kh[idx] = s;
}

// ---------------------------------------------------------------
// Kernel 4: mvs[b,h,d] = (hsum[b,:]·Wv[h*64+d,:]) / len ; kvs from kh similarly
__global__ __launch_bounds__(256) void pass2_kernel(
    const float* __restrict__ Wv, const float* __restrict__ hs,
    const float* __restrict__ kh, const float* __restrict__ il,
    float* __restrict__ mvs, float* __restrict__ kvs)
{
    int u = blockIdx.x * 256 + threadIdx.x;        // 4 blocks -> 1024 = H*DH
    int h = u >> 6, d = u & 63;
    float mv[B_], kv[B_];
#pragma unroll
    for (int b = 0; b < B_; ++b) { mv[b] = 0.f; kv[b] = 0.f; }
    const float* wrow = Wv + (size_t)(h * DH_ + d) * HID_;
    for (int i = 0; i < HID_; i += 4) {
        float4 w = *(const float4*)(wrow + i);
#pragma unroll
        for (int b = 0; b < B_; ++b) {
            float4 s4 = *(const float4*)(hs + b * HID_ + i);
            float4 k4 = *(const float4*)(kh + (size_t)(b * H_ + h) * HID_ + i);
            mv[b] += w.x * s4.x + w.y * s4.y + w.z * s4.z + w.w * s4.w;
            kv[b] += w.x * k4.x + w.y * k4.y + w.z * k4.z + w.w * k4.w;
        }
    }
#pragma unroll
    for (int b = 0; b < B_; ++b) {
        float inv = il[b];
        mvs[(b * H_ + h) * DH_ + d] = mv[b] * inv;
        kvs[(b * H_ + h) * DH_ + d] = kv[b] * inv;
    }
}

// ---------------------------------------------------------------
// Kernel 5: stream the output: out[b,s,h*64+d] = qs[b,s,h]*mvs + kvs
__global__ __launch_bounds__(256) void pass3_kernel(
    const float* __restrict__ qs, const float* __restrict__ mvs,
    const float* __restrict__ kvs, float* __restrict__ out)
{
    const int b = blockIdx.x >> 7;            // 1024 blocks: 8 b x 128 tiles of 32 rows
    const int srow0 = (blockIdx.x & 127) * 32;
    const int t = threadIdx.x;
    const int h = t >> 4;                     // col = t*4 = h*64 + d4
    float4 mv4 = *(const float4*)(mvs + (b * H_ + h) * DH_ + (t & 15) * 4);
    float4 kv4 = *(const float4*)(kvs + (b * H_ + h) * DH_ + (t & 15) * 4);
#pragma unroll 4
    for (int r = 0; r < 32; ++r) {
        size_t srow = (size_t)b * S_ + srow0 + r;
        float q = qs[srow * H_ + h];
        float4 o;
        o.x = q * mv4.x + kv4.x; o.y = q * mv4.y + kv4.y;
        o.z = q * mv4.z + kv4.z; o.w = q * mv4.w + kv4.w;
        *(float4*)(out + srow * (H_ * DH_) + t * 4) = o;
    }
}

// ---------------------------------------------------------------
extern "C" void kernel_launch(void* const* d_in, const int* in_sizes, int n_in,
                              void* d_out, int out_size, void* d_ws, size_t ws_size,
                              hipStream_t stream) {
    const float* hidden = (const float*)d_in[0];
    const float* mask   = (const float*)d_in[1];
    const float* Wq     = (const float*)d_in[2];
    const float* Wk     = (const float*)d_in[3];
    const float* Wv     = (const float*)d_in[4];
    const float* aw     = (const float*)d_in[5];
    float* out = (float*)d_out;
    float* ws  = (float*)d_ws;

    weff_kernel   <<<64,   256, 0, stream>>>(Wq, Wk, aw, ws + OFF_WQE, ws + OFF_WKE);
    invlen_kernel <<<B_,   256, 0, stream>>>(mask, ws + OFF_IL);
    pass1_kernel  <<<256,  256, 0, stream>>>(hidden, mask, ws + OFF_WQE, ws + OFF_WKE,
                                             ws + OFF_QS, ws + OFF_HSP, ws + OFF_KHP);
    reduce_hsum_kernel<<<32,  256, 0, stream>>>(ws + OFF_HSP, ws + OFF_HS);
    reduce_kh_kernel  <<<512, 256, 0, stream>>>(ws + OFF_KHP, ws + OFF_KH);
    pass2_kernel  <<<4,    256, 0, stream>>>(Wv, ws + OFF_HS, ws + OFF_KH, ws + OFF_IL,
                                             ws + OFF_MVS, ws + OFF_KVS);
    pass3_kernel  <<<1024, 256, 0, stream>>>(ws + OFF_QS, ws + OFF_MVS, ws + OFF_KVS, out);
}